// MedianFilter2D_85126251807210
// MI455X (gfx1250) — compile-verified
//
#include <hip/hip_runtime.h>
#include <cstdint>

// Problem constants from the reference: x is (8, 512, 512, 8) float32, NHWC.
#define B_ 8
#define H_ 512
#define W_ 512
#define C_ 8

#define TILE 16
#define TP   (TILE + 2)   // 18: tile + 1-pixel reflect halo on each side

__device__ __forceinline__ float4 f4min(const float4 a, const float4 b) {
    return make_float4(fminf(a.x, b.x), fminf(a.y, b.y), fminf(a.z, b.z), fminf(a.w, b.w));
}
__device__ __forceinline__ float4 f4max(const float4 a, const float4 b) {
    return make_float4(fmaxf(a.x, b.x), fmaxf(a.y, b.y), fmaxf(a.z, b.z), fmaxf(a.w, b.w));
}
// compare-exchange: min -> a, max -> b
#define SORT2(a, b) do { float4 _t = f4min(a, b); (b) = f4max(a, b); (a) = _t; } while (0)

__global__ __launch_bounds__(256) void median3x3_reflect_kernel(
        const float* __restrict__ x, float* __restrict__ out) {
    // Two SoA planes (channels 0-3 and 4-7) of 18x18 pixels, float4 each.
    // 16 B per pixel per plane -> 16 consecutive lanes reading ds_load_b128
    // cover exactly one 256 B LDS bank row: conflict-free.
    __shared__ __align__(16) float4 plane[2][TP * TP];   // 10368 B total

    const int tid = threadIdx.x;
    const int w0  = blockIdx.x * TILE;
    const int h0  = blockIdx.y * TILE;
    const int b   = blockIdx.z;

    // ---- Stage halo tile into LDS with CDNA5 async global->LDS copies ----
    // 18*18 pixels * 2 float4 quads = 648 b128 transfers, per-lane addressed
    // so each lane can apply the reflect-padding index map at the border.
    const int NQ = TP * TP * 2;
    for (int i = tid; i < NQ; i += 256) {
        const int p  = i >> 1;
        const int hf = i & 1;          // channel-half: 0 -> c0..3, 1 -> c4..7
        const int r  = p / TP;
        const int c  = p - r * TP;
        int hh = h0 + r - 1;
        hh = (hh < 0) ? -hh : ((hh >= H_) ? (2 * H_ - 2 - hh) : hh);
        int ww = w0 + c - 1;
        ww = (ww < 0) ? -ww : ((ww >= W_) ? (2 * W_ - 2 - ww) : ww);
        const uint64_t gaddr =
            (uint64_t)(uintptr_t)(x + ((((size_t)b * H_ + hh) * W_ + ww) * C_ + hf * 4));
        // generic addr of LDS object: low 32 bits are the LDS byte offset
        const unsigned laddr = (unsigned)(uintptr_t)(&plane[hf][p]);
        asm volatile("global_load_async_to_lds_b128 %0, %1, off"
                     :: "v"(laddr), "v"(gaddr)
                     : "memory");
    }
    asm volatile("s_wait_asynccnt 0x0" ::: "memory");
    __syncthreads();

    // ---- Each thread: one pixel, 8 channels as two float4 halves ----
    const int r = tid >> 4;   // 0..15
    const int c = tid & 15;   // 0..15
    float* op = out + (((size_t)b * H_ + (h0 + r)) * W_ + (w0 + c)) * C_;

    #pragma unroll
    for (int hf = 0; hf < 2; ++hf) {
        const float4* pl = plane[hf];
        // 3x3 neighborhood of pixel (r+1, c+1) in the padded tile
        float4 v0 = pl[(r + 0) * TP + (c + 0)];
        float4 v1 = pl[(r + 0) * TP + (c + 1)];
        float4 v2 = pl[(r + 0) * TP + (c + 2)];
        float4 v3 = pl[(r + 1) * TP + (c + 0)];
        float4 v4 = pl[(r + 1) * TP + (c + 1)];
        float4 v5 = pl[(r + 1) * TP + (c + 2)];
        float4 v6 = pl[(r + 2) * TP + (c + 0)];
        float4 v7 = pl[(r + 2) * TP + (c + 1)];
        float4 v8 = pl[(r + 2) * TP + (c + 2)];

        // McGuire 19-exchange median-of-9 network; median ends in v4.
        SORT2(v1, v2); SORT2(v4, v5); SORT2(v7, v8);
        SORT2(v0, v1); SORT2(v3, v4); SORT2(v6, v7);
        SORT2(v1, v2); SORT2(v4, v5); SORT2(v7, v8);
        SORT2(v0, v3); SORT2(v5, v8); SORT2(v4, v7);
        SORT2(v3, v6); SORT2(v1, v4); SORT2(v2, v5);
        SORT2(v4, v7); SORT2(v4, v2); SORT2(v6, v4);
        SORT2(v4, v2);

        *(float4*)(op + hf * 4) = v4;
    }
}

extern "C" void kernel_launch(void* const* d_in, const int* in_sizes, int n_in,
                              void* d_out, int out_size, void* d_ws, size_t ws_size,
                              hipStream_t stream) {
    (void)in_sizes; (void)n_in; (void)d_ws; (void)ws_size; (void)out_size;
    const float* x = (const float*)d_in[0];
    float* out     = (float*)d_out;
    dim3 grid(W_ / TILE, H_ / TILE, B_);  // 32 x 32 x 8 = 8192 blocks
    dim3 block(256);
    median3x3_reflect_kernel<<<grid, block, 0, stream>>>(x, out);
}